// Attention_28973849379558
// MI455X (gfx1250) — compile-verified
//
#include <hip/hip_runtime.h>
#include <hip/hip_bf16.h>

typedef float v2f __attribute__((ext_vector_type(2)));
typedef float v8f __attribute__((ext_vector_type(8)));
typedef int   v4i __attribute__((vector_size(16)));

#define B_SZ    1024
#define S_SZ    1024
#define D       128     // D_T == D_S == HIDDEN
#define NWAVE   8
#define ASTRIDE 132     // 128 + 4 floats pad -> conflict-free WMMA A-operand fetch

// ---- LDS partition (floats) ----
#define SM_UB     0                                   // 32*128*4 = 16384 : U_a quad-interleaved
#define SM_ASTRIP (SM_UB + 32*128*4)                  // 8 waves * 2 bufs * 16*132 = 33792
#define SM_HTP    (SM_ASTRIP + NWAVE*2*16*ASTRIDE)    // 128
#define SM_HTL    (SM_HTP + 128)                      // 128
#define SM_VA     (SM_HTL + 128)                      // 128
#define SM_WM     (SM_VA + 128)                       // 8
#define SM_WL     (SM_WM + 8)                         // 8
#define SM_WCTX   (SM_WL + 8)                         // 8*128
#define SM_CC     (SM_WCTX + NWAVE*128)               // 256
#define SM_ATTN   (SM_CC + 256)                       // 128
#define SM_STATS  (SM_ATTN + 128)                     // 2
#define SM_TOTAL  (SM_STATS + 2)                      // ~203 KB total

typedef __attribute__((address_space(1))) v4i GV4;    // global 16B vector
typedef __attribute__((address_space(3))) v4i LV4;    // LDS    16B vector

// ---- CDNA5 async memory->LDS copy (ASYNCcnt path), 16B per lane ----
__device__ __forceinline__ void async_copy_b128(const float* gsrc, float* ldst) {
#if __has_builtin(__builtin_amdgcn_global_load_async_to_lds_b128)
    __builtin_amdgcn_global_load_async_to_lds_b128(
        (GV4*)(uintptr_t)gsrc,
        (LV4*)(unsigned int)(uintptr_t)ldst,   // flat LDS addr low 32b == LDS byte offset
        0, 0);
#else
    asm volatile("global_load_async_to_lds_b128 %0, %1, off"
                 :: "v"((unsigned int)(uintptr_t)ldst), "v"(gsrc) : "memory");
#endif
}

#if __has_builtin(__builtin_amdgcn_s_wait_asynccnt)
#define ASYNC_WAIT(n) do { __builtin_amdgcn_s_wait_asynccnt(n); asm volatile("" ::: "memory"); } while (0)
#else
#define ASYNC_WAIT(n) asm volatile("s_wait_asynccnt %0" :: "i"(n) : "memory")
#endif

__device__ __forceinline__ float shfl_xor32(float v, int mask) {
    int lane = (int)(threadIdx.x & 31u);
    int r = __builtin_amdgcn_ds_bpermute(((lane ^ mask) << 2), __float_as_int(v));
    return __int_as_float(r);
}
__device__ __forceinline__ float shfl_get32(float v, int src) {
    int r = __builtin_amdgcn_ds_bpermute((src << 2), __float_as_int(v));
    return __int_as_float(r);
}

__global__ __launch_bounds__(256)
void attn_fused_kernel(const float* __restrict__ h_t, const float* __restrict__ h_s,
                       const float* __restrict__ W_a, const float* __restrict__ U_a,
                       const float* __restrict__ V_a, const float* __restrict__ W_c,
                       const float* __restrict__ b_c, const float* __restrict__ gma,
                       const float* __restrict__ bta, float* __restrict__ out)
{
    extern __shared__ float sm[];
    float* UB    = sm + SM_UB;
    float* htp   = sm + SM_HTP;
    float* htl   = sm + SM_HTL;
    float* Vs    = sm + SM_VA;
    float* wm    = sm + SM_WM;
    float* wl    = sm + SM_WL;
    float* wctx  = sm + SM_WCTX;
    float* cc    = sm + SM_CC;
    float* attnv = sm + SM_ATTN;
    float* stats = sm + SM_STATS;

    const int b    = blockIdx.x;
    const int t    = (int)threadIdx.x;
    const int wave = t >> 5;
    const int lane = t & 31;
    const int half = lane >> 4;
    const int idx  = lane & 15;

    const float* hsb = h_s + (size_t)b * S_SZ * D;
    float* Aw0 = sm + SM_ASTRIP + wave * (2 * 16 * ASTRIDE);
    float* Aw1 = Aw0 + 16 * ASTRIDE;

    // ---- kick off async DMA of chunk 0 strip immediately ----
    {
        const int s0 = wave * 16;
#pragma unroll 4
        for (int r = 0; r < 16; ++r)
            async_copy_b128(hsb + (size_t)(s0 + r) * D + lane * 4,
                            Aw0 + r * ASTRIDE + lane * 4);
    }

    // ---- stage small operands (overlaps with DMA) ----
    if (t < 128) { htl[t] = h_t[b * D + t]; Vs[t] = V_a[t]; }
    // U_a -> quad-interleaved LDS: UB[(k>>2)*512 + h*4 + (k&3)]
    for (int i = t; i < D * D; i += 256) {
        int k = i >> 7, h = i & 127;
        UB[(k >> 2) * 512 + h * 4 + (k & 3)] = U_a[i];
    }
    __syncthreads();

    // ---- ht_proj = h_t[b] @ W_a (W_a is L2-resident, coalesced over h) ----
    if (t < 128) {
        float acc = 0.f;
#pragma unroll 8
        for (int d = 0; d < D; ++d) acc = fmaf(htl[d], W_a[d * D + t], acc);
        htp[t] = acc;
    }
    __syncthreads();

    // ---- main streaming loop: online softmax over S, h_s read exactly once ----
    float m = -__builtin_inff(), l = 0.f;
    float ctx0 = 0.f, ctx1 = 0.f, ctx2 = 0.f, ctx3 = 0.f;

    for (int c = 0; c < 8; ++c) {
        float* Acur = (c & 1) ? Aw1 : Aw0;
        float* Anxt = (c & 1) ? Aw0 : Aw1;

        if (c < 7) {
            // issue async DMA for next chunk, then wait until only those 16
            // remain outstanding (in-order completion => current chunk landed)
            const int s1 = (c + 1) * 128 + wave * 16;
#pragma unroll 4
            for (int r = 0; r < 16; ++r)
                async_copy_b128(hsb + (size_t)(s1 + r) * D + lane * 4,
                                Anxt + r * ASTRIDE + lane * 4);
            ASYNC_WAIT(16);
        } else {
            ASYNC_WAIT(0);
        }

        float rowsum[8] = {0.f,0.f,0.f,0.f,0.f,0.f,0.f,0.f};
        // energy tile = tanh(strip @ U_a + htp); score = energy @ V_a
        for (int h0 = 0; h0 < 8; ++h0) {
            v8f acc = {};
            const int hcol = h0 * 16 + idx;
#pragma unroll
            for (int k = 0; k < 128; k += 4) {
                v2f a  = *(const v2f*)(Acur + idx * ASTRIDE + k + 2 * half);
                v2f bb = *(const v2f*)(UB + (k >> 2) * 512 + hcol * 4 + 2 * half);
                acc = __builtin_amdgcn_wmma_f32_16x16x4_f32(
                        false, a, false, bb, (short)0, acc, false, false);
            }
            const float htv = htp[hcol];
            const float vav = Vs[hcol];
#pragma unroll
            for (int r = 0; r < 8; ++r)
                rowsum[r] = fmaf(tanhf(acc[r] + htv), vav, rowsum[r]);
        }
        // reduce over the 16 lanes of each half (N dimension)
#pragma unroll
        for (int r = 0; r < 8; ++r) {
            float v = rowsum[r];
            v += shfl_xor32(v, 1); v += shfl_xor32(v, 2);
            v += shfl_xor32(v, 4); v += shfl_xor32(v, 8);
            rowsum[r] = v;
        }
        // gather the strip's 16 row scores into every lane
        float sc[16];
#pragma unroll
        for (int r = 0; r < 8; ++r) {
            sc[r]     = shfl_get32(rowsum[r], 0);   // rows 0..7  (half 0)
            sc[r + 8] = shfl_get32(rowsum[r], 16);  // rows 8..15 (half 1)
        }
        // online softmax update against the LDS-resident strip
        float mloc = sc[0];
#pragma unroll
        for (int r = 1; r < 16; ++r) mloc = fmaxf(mloc, sc[r]);
        const float mnew = fmaxf(m, mloc);
        const float resc = __expf(m - mnew);
        l *= resc; ctx0 *= resc; ctx1 *= resc; ctx2 *= resc; ctx3 *= resc;
#pragma unroll
        for (int r = 0; r < 16; ++r) {
            const float w = __expf(sc[r] - mnew);
            l += w;
            const float* row = Acur + r * ASTRIDE;
            ctx0 = fmaf(w, row[lane],      ctx0);
            ctx1 = fmaf(w, row[lane + 32], ctx1);
            ctx2 = fmaf(w, row[lane + 64], ctx2);
            ctx3 = fmaf(w, row[lane + 96], ctx3);
        }
        m = mnew;
    }

    // ---- flash-combine the 8 waves ----
    if (lane == 0) { wm[wave] = m; wl[wave] = l; }
    wctx[wave * 128 + lane]      = ctx0;
    wctx[wave * 128 + lane + 32] = ctx1;
    wctx[wave * 128 + lane + 64] = ctx2;
    wctx[wave * 128 + lane + 96] = ctx3;
    __syncthreads();

    if (t < 128) {
        float M = wm[0];
#pragma unroll
        for (int w = 1; w < NWAVE; ++w) M = fmaxf(M, wm[w]);
        float L = 0.f, C = 0.f;
#pragma unroll
        for (int w = 0; w < NWAVE; ++w) {
            const float e = __expf(wm[w] - M);
            L = fmaf(wl[w], e, L);
            C = fmaf(wctx[w * 128 + t], e, C);
        }
        cc[t]       = C / L;    // context[d]
        cc[128 + t] = htl[t];   // concat with h_t
    }
    __syncthreads();

    // ---- attn_vec = tanh(concat @ W_c + b_c) ----
    if (t < 128) {
        float acc = b_c[t];
#pragma unroll 8
        for (int j = 0; j < 2 * D; ++j) acc = fmaf(cc[j], W_c[j * D + t], acc);
        attnv[t] = tanhf(acc);
    }
    __syncthreads();

    // ---- LayerNorm stats (one wave) ----
    if (t < 32) {
        float a0 = attnv[t], a1 = attnv[t + 32], a2 = attnv[t + 64], a3 = attnv[t + 96];
        float s = a0 + a1 + a2 + a3;
        float q = a0 * a0 + a1 * a1 + a2 * a2 + a3 * a3;
        for (int msk = 1; msk < 32; msk <<= 1) { s += shfl_xor32(s, msk); q += shfl_xor32(q, msk); }
        if (t == 0) {
            const float mu  = s * (1.f / 128.f);
            const float var = q * (1.f / 128.f) - mu * mu;
            stats[0] = mu;
            stats[1] = rsqrtf(var + 1e-3f);
        }
    }
    __syncthreads();

    if (t < 128)
        out[b * D + t] = (attnv[t] - stats[0]) * stats[1] * gma[t] + bta[t];
}

extern "C" void kernel_launch(void* const* d_in, const int* in_sizes, int n_in,
                              void* d_out, int out_size, void* d_ws, size_t ws_size,
                              hipStream_t stream) {
    (void)in_sizes; (void)n_in; (void)out_size; (void)d_ws; (void)ws_size;
    const float* h_t = (const float*)d_in[0];
    const float* h_s = (const float*)d_in[1];
    const float* W_a = (const float*)d_in[2];
    const float* U_a = (const float*)d_in[3];
    const float* V_a = (const float*)d_in[4];
    const float* W_c = (const float*)d_in[5];
    const float* b_c = (const float*)d_in[6];
    const float* gma = (const float*)d_in[7];
    const float* bta = (const float*)d_in[8];
    float* out = (float*)d_out;

    const size_t shmem = (size_t)SM_TOTAL * sizeof(float);  // ~203 KB (<=320 KB/WGP)
    (void)hipFuncSetAttribute((const void*)attn_fused_kernel,
                              hipFuncAttributeMaxDynamicSharedMemorySize, (int)shmem);
    attn_fused_kernel<<<B_SZ, 256, shmem, stream>>>(h_t, h_s, W_a, U_a, V_a,
                                                    W_c, b_c, gma, bta, out);
}